// MaskedMultiHeadAttention_68539088110139
// MI455X (gfx1250) — compile-verified
//
#include <hip/hip_runtime.h>
#include <math.h>

// Problem dims (fixed by reference)
#define B_   8
#define L_   1024
#define D_   512
#define H_   8
#define DH_  64
#define EPS_ 1e-5f

typedef __attribute__((ext_vector_type(2))) float v2f;
typedef __attribute__((ext_vector_type(8))) float v8f;

// Full-precision CDNA5 matrix op: D(16x16,f32) = A(16x4,f32) * B(4x16,f32) + C
__device__ __forceinline__ v8f wmma4(v2f a, v2f b, v8f c) {
  return __builtin_amdgcn_wmma_f32_16x16x4_f32(
      /*neg_a=*/false, a, /*neg_b=*/false, b,
      /*c_mod=*/(short)0, c, /*reuse_a=*/false, /*reuse_b=*/false);
}

// Fragment lane mapping (ISA 7.12.2, 32-bit operands, wave32):
//   A 16x4 : lane&15 = M row; VGPR j holds K = (lane>=16 ? 2 : 0) + j
//   B 4x16 : lane&15 = N col; VGPR j holds K = (lane>=16 ? 2 : 0) + j
//   C 16x16: lane&15 = N col; VGPR i holds M = i + (lane>=16 ? 8 : 0)

// ---------------------------------------------------------------------------
// Kernel 1: q = query@Wq^T, k = key@Wk^T, v = value@Wv^T  -> [B,H,L,DH] in ws
// One wave per 16x16 output tile; K=512 -> 128 WMMA per tile.
// ---------------------------------------------------------------------------
__global__ __launch_bounds__(256) void qkv_proj_kernel(
    const float* __restrict__ query, const float* __restrict__ key,
    const float* __restrict__ value, const float* __restrict__ Wq,
    const float* __restrict__ Wk, const float* __restrict__ Wv,
    float* __restrict__ qo, float* __restrict__ ko, float* __restrict__ vo) {
  const int lane = threadIdx.x & 31;
  const int wave = threadIdx.x >> 5;
  const int wglobal = blockIdx.x * 8 + wave;
  const int TILES = (B_ * L_ / 16) * (D_ / 16);  // 512 * 32 = 16384 per matrix
  const int mat = wglobal / TILES;               // 0=q 1=k 2=v
  const int t = wglobal % TILES;
  const int mtile = t / (D_ / 16);
  const int ntile = t % (D_ / 16);

  const float* X = (mat == 0) ? query : (mat == 1) ? key : value;
  const float* W = (mat == 0) ? Wq    : (mat == 1) ? Wk  : Wv;
  float*       O = (mat == 0) ? qo    : (mat == 1) ? ko  : vo;

  const int half = lane >> 4;      // 0 or 1
  const int l15  = lane & 15;
  const int kb   = half * 2;
  const int m = mtile * 16 + l15;  // row of X   (A fragment row)
  const int n = ntile * 16 + l15;  // row of W = output col (B fragment col)

  const float* arow = X + (long)m * D_;
  const float* brow = W + (long)n * D_;  // out = X @ W^T -> B[kd][n] = W[n][kd]

  v8f c = {};
  for (int k = 0; k < D_; k += 4) {
    v2f a = *(const v2f*)(arow + k + kb);
    v2f b = *(const v2f*)(brow + k + kb);
    c = wmma4(a, b, c);
  }

  // scatter C into [B,H,L,DH]
#pragma unroll
  for (int i = 0; i < 8; ++i) {
    int r   = mtile * 16 + i + half * 8;  // global row in [0, B*L)
    int col = ntile * 16 + l15;           // [0, D)
    int bb = r >> 10, l = r & (L_ - 1);
    int h = col >> 6, dh = col & 63;
    O[(((long)(bb * H_ + h) * L_ + l) << 6) + dh] = c[i];
  }
}

// ---------------------------------------------------------------------------
// Kernel 2: per (b, h, 16-row q-tile): scores -> LDS, softmax + masks,
// attention -> d_out, context = attn @ V -> ws.  8 waves / block.
// ---------------------------------------------------------------------------
__global__ __launch_bounds__(256) void attn_kernel(
    const float* __restrict__ qm, const float* __restrict__ km,
    const float* __restrict__ vm, const unsigned char* __restrict__ amask,
    const unsigned char* __restrict__ smask, float* __restrict__ att_out,
    float* __restrict__ ctx) {
  __shared__ float S[16 * L_];  // 64 KB score / attention row-block

  const int lane = threadIdx.x & 31;
  const int wave = threadIdx.x >> 5;
  const int qt = blockIdx.x & (L_ / 16 - 1);  // q tile 0..63
  const int bh = blockIdx.x >> 6;             // b*H + h
  const int b  = bh / H_;
  const int h  = bh % H_;
  const int half = lane >> 4, l15 = lane & 15, kb = half * 2;

  const float* qbase = qm + (long)bh * L_ * DH_;
  const float* kbase = km + (long)bh * L_ * DH_;
  const float* vbase = vm + (long)bh * L_ * DH_;

  // Preload the 16 A-fragments of this q-tile (d = 0..63), reused for all k-tiles
  const int mrow = qt * 16 + l15;
  v2f areg[16];
#pragma unroll
  for (int kk = 0; kk < 16; ++kk)
    areg[kk] = *(const v2f*)(qbase + (long)mrow * DH_ + kk * 4 + kb);

  // ---- scores = (q . k^T) * scale, attn_mask -> -inf, into LDS ----
  for (int kt = wave; kt < L_ / 16; kt += 8) {
    const int ncol = kt * 16 + l15;                 // key sequence position
    const float* brow = kbase + (long)ncol * DH_;   // B[d][ncol] = k[ncol][d]
    v8f c = {};
#pragma unroll
    for (int kk = 0; kk < 16; ++kk) {
      v2f bf = *(const v2f*)(brow + kk * 4 + kb);
      c = wmma4(areg[kk], bf, c);
    }
#pragma unroll
    for (int i = 0; i < 8; ++i) {
      int r  = i + half * 8;         // row within q-tile
      int qr = qt * 16 + r;
      int kc = kt * 16 + l15;
      float val = c[i] * 0.125f;     // DH^-0.5 = 1/8
      if (amask[((long)b * L_ + qr) * L_ + kc]) val = -__builtin_inff();
      S[r * L_ + kc] = val;
    }
  }
  __syncthreads();

  // ---- row softmax (16 threads per row, shuffle reductions, wave32) ----
  {
    const int row = threadIdx.x >> 4;  // 0..15
    const int sub = threadIdx.x & 15;  // 0..15, each covers 64 columns
    const int qr  = qt * 16 + row;
    float mx = -__builtin_inff();
    for (int c0 = sub * 64; c0 < sub * 64 + 64; ++c0)
      mx = fmaxf(mx, S[row * L_ + c0]);
#pragma unroll
    for (int off = 8; off >= 1; off >>= 1)
      mx = fmaxf(mx, __shfl_xor(mx, off, 16));
    float sum = 0.f;
    for (int c0 = sub * 64; c0 < sub * 64 + 64; ++c0)
      sum += __expf(S[row * L_ + c0] - mx);
#pragma unroll
    for (int off = 8; off >= 1; off >>= 1)
      sum += __shfl_xor(sum, off, 16);
    float inv = 1.0f / sum;
    for (int c0 = sub * 64; c0 < sub * 64 + 64; ++c0) {
      float a = __expf(S[row * L_ + c0] - mx) * inv;
      if (smask[((long)b * L_ + qr) * L_ + c0]) a = 0.f;  // softmax_mask
      S[row * L_ + c0] = a;
      att_out[((long)bh * L_ + qr) * L_ + c0] = a;        // [B,H,L,L] output
    }
  }
  __syncthreads();

  // ---- context = attn(16x1024, LDS) @ V(1024x64): 4 n-tiles, waves 0..3 ----
  if (wave < 4) {
    const int nt = wave;
    const int ncol = nt * 16 + l15;  // dh
    v8f c = {};
    for (int k = 0; k < L_; k += 4) {
      v2f a = *(const v2f*)(&S[l15 * L_ + k + kb]);  // A row = l15, K = k+kb..+1
      v2f bf;
      bf.x = vbase[(long)(k + kb)     * DH_ + ncol]; // B[kd][ncol] = v[kd][ncol]
      bf.y = vbase[(long)(k + kb + 1) * DH_ + ncol];
      c = wmma4(a, bf, c);
    }
#pragma unroll
    for (int i = 0; i < 8; ++i) {
      int r  = i + half * 8;
      int qr = qt * 16 + r;
      // ctx layout [B, L, D] with heads merged: col = h*64 + dh
      ctx[((long)b * L_ + qr) * D_ + h * DH_ + nt * 16 + l15] = c[i];
    }
  }
}

// ---------------------------------------------------------------------------
// Kernel 3: x = ctx @ Wo^T + residual(query); LayerNorm rows -> d_out
// One block = 16 rows; 8 waves cover 32 n-tiles; K=512.
// ---------------------------------------------------------------------------
__global__ __launch_bounds__(256) void out_ln_kernel(
    const float* __restrict__ ctx, const float* __restrict__ Wo,
    const float* __restrict__ query, float* __restrict__ out) {
  __shared__ float X[16 * 516];  // padded row stride vs 64 LDS banks

  const int lane = threadIdx.x & 31;
  const int wave = threadIdx.x >> 5;
  const int half = lane >> 4, l15 = lane & 15, kb = half * 2;
  const int mtile = blockIdx.x;          // 0..511
  const int m = mtile * 16 + l15;

  for (int nt = wave; nt < D_ / 16; nt += 8) {
    const int n = nt * 16 + l15;
    const float* arow = ctx + (long)m * D_;
    const float* brow = Wo + (long)n * D_;  // x = ctx @ Wo^T
    v8f c = {};
    for (int k = 0; k < D_; k += 4) {
      v2f a = *(const v2f*)(arow + k + kb);
      v2f b = *(const v2f*)(brow + k + kb);
      c = wmma4(a, b, c);
    }
#pragma unroll
    for (int i = 0; i < 8; ++i) {
      int r   = i + half * 8;
      int gr  = mtile * 16 + r;
      int col = nt * 16 + l15;
      X[r * 516 + col] = c[i] + query[(long)gr * D_ + col];  // + residual
    }
  }
  __syncthreads();

  // LayerNorm: 16 threads per row, shuffle reductions
  const int row = threadIdx.x >> 4, sub = threadIdx.x & 15;
  float s = 0.f, sq = 0.f;
  for (int c0 = sub * 32; c0 < sub * 32 + 32; ++c0) {
    float xv = X[row * 516 + c0];
    s += xv;
    sq += xv * xv;
  }
#pragma unroll
  for (int off = 8; off >= 1; off >>= 1) {
    s  += __shfl_xor(s, off, 16);
    sq += __shfl_xor(sq, off, 16);
  }
  float mean = s * (1.0f / D_);
  float var  = sq * (1.0f / D_) - mean * mean;
  float rstd = rsqrtf(var + EPS_);
  int gr = mtile * 16 + row;
  for (int c0 = sub * 32; c0 < sub * 32 + 32; ++c0)
    out[(long)gr * D_ + c0] = (X[row * 516 + c0] - mean) * rstd;
}

// ---------------------------------------------------------------------------
extern "C" void kernel_launch(void* const* d_in, const int* in_sizes, int n_in,
                              void* d_out, int out_size, void* d_ws, size_t ws_size,
                              hipStream_t stream) {
  const float* query = (const float*)d_in[0];
  const float* key   = (const float*)d_in[1];
  const float* value = (const float*)d_in[2];
  const unsigned char* amask = (const unsigned char*)d_in[3];  // jax bool = u8
  const unsigned char* smask = (const unsigned char*)d_in[4];
  const float* Wq = (const float*)d_in[5];
  const float* Wk = (const float*)d_in[6];
  const float* Wv = (const float*)d_in[7];
  const float* Wo = (const float*)d_in[8];

  float* out = (float*)d_out;                       // [B,L,D]
  float* att_out = out + (size_t)B_ * L_ * D_;      // [B,H,L,L]

  float* ws = (float*)d_ws;
  const size_t N = (size_t)B_ * L_ * D_;            // 4M floats each
  float* qbuf = ws;
  float* kbuf = ws + N;
  float* vbuf = ws + 2 * N;
  float* ctx  = ws + 3 * N;

  // 3 matrices * 512 * 32 tiles / 8 waves-per-block = 6144 blocks
  qkv_proj_kernel<<<6144, 256, 0, stream>>>(query, key, value, Wq, Wk, Wv,
                                            qbuf, kbuf, vbuf);
  // one block per (b, h, q-tile): 8*8*64 = 4096
  attn_kernel<<<4096, 256, 0, stream>>>(qbuf, kbuf, vbuf, amask, smask,
                                        att_out, ctx);
  // one block per 16 output rows: 8192/16 = 512
  out_ln_kernel<<<512, 256, 0, stream>>>(ctx, Wo, query, out);
}